// GCN_50680614093544
// MI455X (gfx1250) — compile-verified
//
#include <hip/hip_runtime.h>

typedef __attribute__((ext_vector_type(2))) float v2f;
typedef __attribute__((ext_vector_type(8))) float v8f;

#define BN_EPS 1e-5f

// =====================================================================
// GEMM: Y[M x NOUT] = A[M x 64] @ W[64 x NOUT] (+bias) (+relu)
// fp32 WMMA 16x16x4. One wave owns a 16-node x 16-feature tile.
// NOUT=64: 8 waves = 2 node tiles x 4 feat tiles -> 32 nodes/block.
// NOUT=128: 8 waves = 1 node tile x 8 feat tiles -> 16 nodes/block.
// =====================================================================
template <int NOUT, bool BIAS, bool RELU>
__global__ __launch_bounds__(256) void gemm_k64_wmma(
    const float* __restrict__ A, const float* __restrict__ W,
    const float* __restrict__ bias, float* __restrict__ Y) {
  __shared__ float Ws[64 * NOUT];
  const int tid = threadIdx.x;
  for (int i = tid; i < 64 * NOUT; i += 256) Ws[i] = W[i];
  __syncthreads();

  const int wave = tid >> 5;
  const int lane = tid & 31;
  constexpr int FT = NOUT / 16;                 // feature tiles per row
  constexpr int NODES_PER_BLOCK = 16 * (8 / FT);
  const int featTile = wave % FT;
  const int nodeTile = wave / FT;
  const int node_base = blockIdx.x * NODES_PER_BLOCK + nodeTile * 16;
  const int m = lane & 15;
  const int n = featTile * 16 + m;
  const int koff = (lane < 16) ? 0 : 2;         // ISA A/B f32 fragment layout

  const float* Arow = A + (node_base + m) * 64;

  v8f c;
  const float bv = BIAS ? bias[n] : 0.0f;
#pragma unroll
  for (int g = 0; g < 8; ++g) c[g] = bv;

#pragma unroll
  for (int kb = 0; kb < 64; kb += 4) {
    v2f a = *(const v2f*)(Arow + kb + koff);    // A[m][kb+koff .. +1]
    v2f b;
    b.x = Ws[(kb + koff) * NOUT + n];           // B[k][n]
    b.y = Ws[(kb + koff + 1) * NOUT + n];
    c = __builtin_amdgcn_wmma_f32_16x16x4_f32(false, a, false, b, (short)0, c,
                                              false, false);
  }

  const int mrow = node_base + ((lane < 16) ? 0 : 8);
#pragma unroll
  for (int g = 0; g < 8; ++g) {
    float v = c[g];
    if (RELU) v = fmaxf(v, 0.0f);
    Y[(mrow + g) * NOUT + n] = v;
  }
}

// =====================================================================
// Degree / norm precompute
// =====================================================================
__global__ void degree_kernel(const int* __restrict__ dst,
                              float* __restrict__ deg, int E) {
  int e = blockIdx.x * blockDim.x + threadIdx.x;
  if (e < E) unsafeAtomicAdd(&deg[dst[e]], 1.0f);
}

__global__ void dinv_kernel(float* __restrict__ deg, int N) {
  int n = blockIdx.x * blockDim.x + threadIdx.x;
  if (n < N) deg[n] = rsqrtf(deg[n] + 1.0f);   // in-place: deg -> deg^{-1/2}
}

__global__ void norm_kernel(const int* __restrict__ src,
                            const int* __restrict__ dst,
                            const float* __restrict__ dinv,
                            float* __restrict__ norm, int E) {
  int e = blockIdx.x * blockDim.x + threadIdx.x;
  if (e < E) norm[e] = dinv[src[e]] * dinv[dst[e]];
}

// agg[n][f] = h[n][f] * dinv[n]^2 + bias[f]   (self-loop term + layer bias)
__global__ void agg_init_kernel(const float* __restrict__ h,
                                const float* __restrict__ dinv,
                                const float* __restrict__ bs,
                                float* __restrict__ agg, int total) {
  int idx = blockIdx.x * blockDim.x + threadIdx.x;
  if (idx >= total) return;
  int node = idx >> 6, f = idx & 63;
  float di = dinv[node];
  agg[idx] = h[idx] * di * di + bs[f];
}

// one wave per edge, 2 features per lane -> 64 features
__global__ __launch_bounds__(256) void scatter_kernel(
    const int* __restrict__ src, const int* __restrict__ dst,
    const float* __restrict__ norm, const float* __restrict__ h,
    float* __restrict__ agg, int E) {
  int gid = blockIdx.x * blockDim.x + threadIdx.x;
  int e = gid >> 5;
  int lane = threadIdx.x & 31;
  if (e >= E) return;
  int s = src[e], d = dst[e];
  float w = norm[e];
  float2 hv = *(const float2*)(h + (size_t)s * 64 + lane * 2);
  float* out = agg + (size_t)d * 64 + lane * 2;
  unsafeAtomicAdd(out, hv.x * w);
  unsafeAtomicAdd(out + 1, hv.y * w);
}

// =====================================================================
// BatchNorm: per-channel stats over N nodes (64 blocks, one per channel)
// =====================================================================
__global__ __launch_bounds__(256) void bn_stats_kernel(
    const float* __restrict__ h, float* __restrict__ stats, int N) {
  const int c = blockIdx.x;
  float s = 0.f, s2 = 0.f;
  for (int n = threadIdx.x; n < N; n += 256) {
    float v = h[(size_t)n * 64 + c];
    s += v;
    s2 += v * v;
  }
  __shared__ float r0[256], r1[256];
  r0[threadIdx.x] = s;
  r1[threadIdx.x] = s2;
  __syncthreads();
  for (int off = 128; off > 0; off >>= 1) {
    if (threadIdx.x < off) {
      r0[threadIdx.x] += r0[threadIdx.x + off];
      r1[threadIdx.x] += r1[threadIdx.x + off];
    }
    __syncthreads();
  }
  if (threadIdx.x == 0) {
    float mean = r0[0] / (float)N;
    stats[c] = mean;
    stats[64 + c] = r1[0] / (float)N - mean * mean;  // biased var (jnp.var)
  }
}

__global__ void bn_apply_kernel(float* __restrict__ h,
                                const float* __restrict__ stats,
                                const float* __restrict__ gamma,
                                const float* __restrict__ beta, int relu,
                                int total) {
  int idx = blockIdx.x * blockDim.x + threadIdx.x;
  if (idx >= total) return;
  int f = idx & 63;
  float v = (h[idx] - stats[f]) * rsqrtf(stats[64 + f] + BN_EPS);
  v = gamma[f] * v + beta[f];
  if (relu) v = fmaxf(v, 0.0f);
  h[idx] = v;
}

// =====================================================================
// out[n] = dot(h1[n][0..127], W2) + b2 ; one wave per node, 4 feats/lane
// =====================================================================
__global__ __launch_bounds__(256) void out_dot_kernel(
    const float* __restrict__ H1, const float* __restrict__ W2,
    const float* __restrict__ b2, float* __restrict__ out, int N) {
  int wid = (blockIdx.x * blockDim.x + threadIdx.x) >> 5;
  int lane = threadIdx.x & 31;
  if (wid >= N) return;
  float4 h = *(const float4*)(H1 + (size_t)wid * 128 + lane * 4);
  float4 w = *(const float4*)(W2 + lane * 4);
  float s = h.x * w.x + h.y * w.y + h.z * w.z + h.w * w.w;
#pragma unroll
  for (int off = 16; off > 0; off >>= 1) s += __shfl_down(s, off, 32);
  if (lane == 0) out[wid] = s + b2[0];
}

// =====================================================================
// Host-side orchestration
// =====================================================================
extern "C" void kernel_launch(void* const* d_in, const int* in_sizes, int n_in,
                              void* d_out, int out_size, void* d_ws,
                              size_t ws_size, hipStream_t stream) {
  const float* X = (const float*)d_in[0];
  const int* edge = (const int*)d_in[1];
  const float* W_in = (const float*)d_in[2];
  const float* b_in = (const float*)d_in[3];
  const float* Ws = (const float*)d_in[4];
  const float* bs = (const float*)d_in[5];
  const float* gammas = (const float*)d_in[6];
  const float* betas = (const float*)d_in[7];
  const float* W1 = (const float*)d_in[8];
  const float* b1 = (const float*)d_in[9];
  const float* W2 = (const float*)d_in[10];
  const float* b2 = (const float*)d_in[11];
  float* out = (float*)d_out;

  const int N = in_sizes[0] / 64;  // 100000
  const int E = in_sizes[1] / 2;   // 1600000
  const int* srcIdx = edge;
  const int* dstIdx = edge + E;

  // workspace layout (floats)
  float* w = (float*)d_ws;
  float* dinv = w;                       // N   (deg, then deg^{-1/2})
  float* normv = dinv + N;               // E
  float* buf0 = normv + E;               // N*64
  float* buf1 = buf0 + (size_t)N * 64;   // N*64
  float* bufH = buf1 + (size_t)N * 64;   // N*64
  float* h1 = bufH + (size_t)N * 64;     // N*128
  float* stats = h1 + (size_t)N * 128;   // 128

  const int T = 256;

  // 1) degree -> dinv
  hipMemsetAsync(dinv, 0, (size_t)N * sizeof(float), stream);
  degree_kernel<<<(E + T - 1) / T, T, 0, stream>>>(dstIdx, dinv, E);
  dinv_kernel<<<(N + T - 1) / T, T, 0, stream>>>(dinv, N);
  norm_kernel<<<(E + T - 1) / T, T, 0, stream>>>(srcIdx, dstIdx, dinv, normv, E);

  // 2) input linear: x = X @ W_in + b_in
  gemm_k64_wmma<64, true, false><<<N / 32, T, 0, stream>>>(X, W_in, b_in, buf0);

  // 3) GCN layers
  float* x = buf0;
  float* other = buf1;
  const int total64 = N * 64;
  for (int i = 0; i < 3; ++i) {
    gemm_k64_wmma<64, false, false>
        <<<N / 32, T, 0, stream>>>(x, Ws + (size_t)i * 64 * 64, nullptr, bufH);
    agg_init_kernel<<<(total64 + T - 1) / T, T, 0, stream>>>(bufH, dinv,
                                                             bs + i * 64,
                                                             other, total64);
    scatter_kernel<<<((size_t)E * 32 + T - 1) / T, T, 0, stream>>>(
        srcIdx, dstIdx, normv, bufH, other, E);
    bn_stats_kernel<<<64, T, 0, stream>>>(other, stats, N);
    bn_apply_kernel<<<(total64 + T - 1) / T, T, 0, stream>>>(
        other, stats, gammas + i * 64, betas + i * 64, (i < 2) ? 1 : 0,
        total64);
    float* t = x;
    x = other;
    other = t;
  }

  // 4) output head
  gemm_k64_wmma<128, true, true><<<N / 16, T, 0, stream>>>(x, W1, b1, h1);
  out_dot_kernel<<<((size_t)N * 32 + T - 1) / T, T, 0, stream>>>(h1, W2, b2,
                                                                 out, N);
}